// SGC_79053168050936
// MI455X (gfx1250) — compile-verified
//
#include <hip/hip_runtime.h>

typedef __attribute__((ext_vector_type(2))) float v2f;
typedef __attribute__((ext_vector_type(8))) float v8f;

#define F_IN   256
#define F_OUT  40
#define NT     48        // F_OUT padded to 3x16 WMMA column tiles
#define WSTR   260       // LDS row stride (floats) for W: 1040B % 256B = 16 -> full bank spread
#define OSTR   52        // LDS row stride (floats) for out tile staging
#define EPB    8         // edges per block in propagation

// ---------------- degree scatter: deg[col] += w ----------------
__global__ __launch_bounds__(256) void sgc_deg(const int* __restrict__ cols,
                                               const float* __restrict__ w,
                                               float* __restrict__ deg, int E) {
  int e = blockIdx.x * 256 + threadIdx.x;
  if (e < E) {
    __hip_atomic_fetch_add(deg + cols[e], w[e], __ATOMIC_RELAXED, __HIP_MEMORY_SCOPE_AGENT);
  }
}

// ---------------- dinv = rsqrt(deg + 1) in place ----------------
__global__ __launch_bounds__(256) void sgc_dinv(float* __restrict__ deg, int N) {
  int i = blockIdx.x * 256 + threadIdx.x;
  if (i < N) {
    float d = deg[i] + 1.0f;                 // +1 self loop weight
    deg[i] = (d > 0.0f) ? rsqrtf(d) : 0.0f;
  }
}

// ---------------- norm[e] = dinv[row]*w*dinv[col] ----------------
__global__ __launch_bounds__(256) void sgc_norm(const int* __restrict__ rows,
                                                const int* __restrict__ cols,
                                                const float* __restrict__ w,
                                                const float* __restrict__ dinv,
                                                float* __restrict__ norm, int E) {
  int e = blockIdx.x * 256 + threadIdx.x;
  if (e < E) norm[e] = dinv[rows[e]] * w[e] * dinv[cols[e]];
}

// ------- edge propagation: dst[col,:] += norm * src[row,:] (atomic scatter) -------
__global__ __launch_bounds__(256) void sgc_prop_edges(const float* __restrict__ src,
                                                      float* __restrict__ dst,
                                                      const int* __restrict__ rows,
                                                      const int* __restrict__ cols,
                                                      const float* __restrict__ norm,
                                                      int E) {
  const int f = threadIdx.x;                  // one feature per thread (coalesced 1KB rows)
  const long ebase = (long)blockIdx.x * EPB;
  for (int i = 0; i < EPB; ++i) {
    long e = ebase + i;
    if (e >= E) return;                       // uniform across block
    int   r  = rows[e];                       // uniform -> scalar loads
    int   c  = cols[e];
    float nv = norm[e];
    float v  = src[(long)r * F_IN + f] * nv;
    __hip_atomic_fetch_add(dst + (long)c * F_IN + f, v,
                           __ATOMIC_RELAXED, __HIP_MEMORY_SCOPE_AGENT);
  }
}

// ------- self loop: dst[i,:] += dinv[i]^2 * src[i,:] (no atomics needed) -------
__global__ __launch_bounds__(256) void sgc_prop_self(const float* __restrict__ src,
                                                     float* __restrict__ dst,
                                                     const float* __restrict__ dinv,
                                                     long total) {
  long g = (long)blockIdx.x * 256 + threadIdx.x;
  if (g < total) {
    float d = dinv[g >> 8];                   // F_IN == 256
    dst[g] = dst[g] + d * d * src[g];
  }
}

// ------- fused GEMM (x2 @ W^T + b) + ReLU + log_softmax via f32 WMMA -------
// block = 256 threads = 8 waves; each wave computes a 16-row x 48-col tile,
// block covers 128 rows. K = 256 consumed in steps of 4 by v_wmma_f32_16x16x4_f32.
__global__ __launch_bounds__(256) void sgc_gemm_lsm(const float* __restrict__ x,
                                                    const float* __restrict__ W,
                                                    const float* __restrict__ bias,
                                                    float* __restrict__ out, int N) {
  __shared__ float smem[NT * WSTR];           // phase1: W staging; phase2: out staging
  __shared__ float sbias[NT];

  const int tid = threadIdx.x;

  // ---- phase 1: stage W (40x256, rows 40..47 zero) into LDS, padded stride ----
  for (int i = tid; i < NT * F_IN; i += 256) {
    int r = i >> 8, c = i & (F_IN - 1);
    smem[r * WSTR + c] = (r < F_OUT) ? W[r * F_IN + c] : 0.0f;
  }
  if (tid < NT) sbias[tid] = (tid < F_OUT) ? bias[tid] : 0.0f;
  __syncthreads();

  const int wave = tid >> 5;
  const int lane = tid & 31;
  const int mloc = lane & 15;                 // A row / B col within tile
  const int kb   = (lane >> 4) << 1;          // K sub-offset: 0 or 2

  long row = (long)blockIdx.x * 128 + wave * 16 + mloc;
  long rowc = (row < N) ? row : (long)(N - 1);          // clamp (invalid rows discarded later)
  const float* ap  = x + rowc * F_IN + kb;
  const float* wp0 = smem + (0 * 16 + mloc) * WSTR + kb;
  const float* wp1 = smem + (1 * 16 + mloc) * WSTR + kb;
  const float* wp2 = smem + (2 * 16 + mloc) * WSTR + kb;

  v8f acc0 = {}, acc1 = {}, acc2 = {};
  for (int k0 = 0; k0 < F_IN; k0 += 4) {
    float2 av = *(const float2*)(ap + k0);               // global_load_b64
    v2f a; a.x = av.x; a.y = av.y;
    float2 b0v = *(const float2*)(wp0 + k0);             // ds_read_b64, conflict-free
    float2 b1v = *(const float2*)(wp1 + k0);
    float2 b2v = *(const float2*)(wp2 + k0);
    v2f b0; b0.x = b0v.x; b0.y = b0v.y;
    v2f b1; b1.x = b1v.x; b1.y = b1v.y;
    v2f b2; b2.x = b2v.x; b2.y = b2v.y;
    acc0 = __builtin_amdgcn_wmma_f32_16x16x4_f32(false, a, false, b0, (short)0, acc0, false, false);
    acc1 = __builtin_amdgcn_wmma_f32_16x16x4_f32(false, a, false, b1, (short)0, acc1, false, false);
    acc2 = __builtin_amdgcn_wmma_f32_16x16x4_f32(false, a, false, b2, (short)0, acc2, false, false);
  }
  __syncthreads();                                       // everyone done reading W from LDS

  // ---- phase 2: stage accumulators (+bias) into LDS as out[128][OSTR] ----
  {
    int mbase = wave * 16 + ((lane >> 4) << 3);          // C/D layout: M = v + 8*(lane>=16)
    int nn = lane & 15;                                  // N = lane % 16
    for (int v = 0; v < 8; ++v) {
      smem[(mbase + v) * OSTR + 0  + nn] = acc0[v] + sbias[0  + nn];
      smem[(mbase + v) * OSTR + 16 + nn] = acc1[v] + sbias[16 + nn];
      smem[(mbase + v) * OSTR + 32 + nn] = acc2[v] + sbias[32 + nn];
    }
  }
  __syncthreads();

  // ---- phase 3: ReLU + log_softmax per row, float4 stores ----
  if (tid < 128) {
    long r = (long)blockIdx.x * 128 + tid;
    if (r < N) {
      const float* rp = smem + tid * OSTR;
      float mx = 0.0f;                                   // relu values are >= 0
      for (int j = 0; j < F_OUT; ++j) mx = fmaxf(mx, fmaxf(rp[j], 0.0f));
      float s = 0.0f;
      for (int j = 0; j < F_OUT; ++j) s += __expf(fmaxf(rp[j], 0.0f) - mx);
      float lg = __logf(s) + mx;
      float4* op = (float4*)(out + r * F_OUT);           // 160B rows: 16B aligned
      for (int j = 0; j < F_OUT / 4; ++j) {
        float4 v;
        v.x = fmaxf(rp[j * 4 + 0], 0.0f) - lg;
        v.y = fmaxf(rp[j * 4 + 1], 0.0f) - lg;
        v.z = fmaxf(rp[j * 4 + 2], 0.0f) - lg;
        v.w = fmaxf(rp[j * 4 + 3], 0.0f) - lg;
        op[j] = v;
      }
    }
  }
}

extern "C" void kernel_launch(void* const* d_in, const int* in_sizes, int n_in,
                              void* d_out, int out_size, void* d_ws, size_t ws_size,
                              hipStream_t stream) {
  const float* x  = (const float*)d_in[0];
  const int*   ei = (const int*)d_in[1];     // edge_index, (2,E) flattened
  const float* ea = (const float*)d_in[2];
  const float* W  = (const float*)d_in[3];
  const float* b  = (const float*)d_in[4];

  const int N = in_sizes[0] / F_IN;
  const int E = in_sizes[2];
  const int* rows = ei;
  const int* cols = ei + E;

  // workspace carve-out (256B-aligned): dinv | norm | x1 | x2
  size_t off = 0;
  auto carve = [&](size_t bytes) -> void* {
    void* p = (void*)((char*)d_ws + off);
    off = (off + bytes + 255) & ~(size_t)255;
    return p;
  };
  float* d_dinv = (float*)carve((size_t)N * 4);
  float* d_norm = (float*)carve((size_t)E * 4);
  float* d_x1   = (float*)carve((size_t)N * F_IN * 4);
  float* d_x2   = (float*)carve((size_t)N * F_IN * 4);
  (void)ws_size; (void)n_in; (void)out_size;

  hipMemsetAsync(d_dinv, 0, (size_t)N * 4, stream);
  hipMemsetAsync(d_x1,   0, (size_t)N * F_IN * 4, stream);
  hipMemsetAsync(d_x2,   0, (size_t)N * F_IN * 4, stream);

  const int eb = (E + 255) / 256;
  const int nb = (N + 255) / 256;

  sgc_deg <<<eb, 256, 0, stream>>>(cols, ea, d_dinv, E);
  sgc_dinv<<<nb, 256, 0, stream>>>(d_dinv, N);
  sgc_norm<<<eb, 256, 0, stream>>>(rows, cols, ea, d_dinv, d_norm, E);

  const int pb = (E + EPB - 1) / EPB;
  const long total = (long)N * F_IN;
  const int sb = (int)((total + 255) / 256);

  // hop 1: x -> x1
  sgc_prop_edges<<<pb, 256, 0, stream>>>(x, d_x1, rows, cols, d_norm, E);
  sgc_prop_self <<<sb, 256, 0, stream>>>(x, d_x1, d_dinv, total);
  // hop 2: x1 -> x2
  sgc_prop_edges<<<pb, 256, 0, stream>>>(d_x1, d_x2, rows, cols, d_norm, E);
  sgc_prop_self <<<sb, 256, 0, stream>>>(d_x1, d_x2, d_dinv, total);

  // fused linear + relu + log_softmax
  sgc_gemm_lsm<<<(N + 127) / 128, 256, 0, stream>>>(d_x2, W, b, (float*)d_out, N);
}